// GeoBleuSinkhornLoss_28028956574253
// MI455X (gfx1250) — compile-verified
//
#include <hip/hip_runtime.h>
#include <hip/hip_bf16.h>
#include <math.h>

typedef __attribute__((ext_vector_type(2))) float v2f;
typedef __attribute__((ext_vector_type(4))) float v4f;
typedef __attribute__((ext_vector_type(8))) float v8f;

#define HH 200
#define WW 200
#define VV 40000      // H*W
#define BB 16
#define TT 96
#define NK 49         // WIN*WIN
#define NGRAMS 5
#define ROWS (BB*TT)  // 1536
#define PITCH 98      // LDS pitch for 96x96 K matrix (bank spread, 8B-aligned rows)

// ---------------------------------------------------------------------------
// Kernel 0: zero the scalar output (d_out is poisoned before timing)
// ---------------------------------------------------------------------------
__global__ void zero_out_kernel(float* out) {
  if (threadIdx.x == 0) out[0] = 0.0f;
}

// ---------------------------------------------------------------------------
// Kernel 1: per-row online softmax stats (max, sum of exp) over V=40000.
// One 256-thread block per (b,t) row; single streaming pass over 245.8 MB.
// ---------------------------------------------------------------------------
__global__ __launch_bounds__(256) void row_stats_kernel(
    const float* __restrict__ logits,
    float* __restrict__ rowmax, float* __restrict__ rowsum) {
  const int row = blockIdx.x;
  const int tid = threadIdx.x;
  const v4f* __restrict__ p4 = (const v4f*)(logits + (size_t)row * VV);
  const int n4 = VV / 4;  // 10000

  float m = -3.4e38f, s = 0.0f;
  for (int i = tid; i < n4; i += 256) {
    __builtin_prefetch(&p4[i + 512], 0, 0);   // global_prefetch_b8 ahead
    v4f x = p4[i];
#pragma unroll
    for (int e = 0; e < 4; ++e) {
      float xv = x[e];
      float nm = fmaxf(m, xv);
      s = s * __expf(m - nm) + __expf(xv - nm);
      m = nm;
    }
  }

  __shared__ float sm[256];
  __shared__ float ss[256];
  sm[tid] = m; ss[tid] = s;
  __syncthreads();
  for (int off = 128; off > 0; off >>= 1) {
    if (tid < off) {
      float m2 = sm[tid + off], s2 = ss[tid + off];
      float nm = fmaxf(sm[tid], m2);
      ss[tid] = ss[tid] * __expf(sm[tid] - nm) + s2 * __expf(m2 - nm);
      sm[tid] = nm;
    }
    __syncthreads();
  }
  if (tid == 0) { rowmax[row] = sm[0]; rowsum[row] = ss[0]; }
}

// ---------------------------------------------------------------------------
// Kernel 2: U[b,t,u] = max(1e-12, (1/Z) * sum_k exp(x[nb(u,k)] - max) * w_k)
// One 96-thread block per (b,t); thread u gathers its 49 neighbors (L2 hits).
// ---------------------------------------------------------------------------
__global__ __launch_bounds__(96) void compute_u_kernel(
    const float* __restrict__ logits, const float* __restrict__ kw,
    const int* __restrict__ tgt, const int* __restrict__ offs,
    const float* __restrict__ rowmax, const float* __restrict__ rowsum,
    float* __restrict__ U) {
  const int bt = blockIdx.x;
  const int b  = bt / TT;
  const int u  = threadIdx.x;

  __shared__ int   s_dy[NK];
  __shared__ int   s_dx[NK];
  __shared__ float s_kw[NK];
  if (u < NK) { s_dy[u] = offs[2*u]; s_dx[u] = offs[2*u + 1]; s_kw[u] = kw[u]; }
  __syncthreads();

  const int t_id = tgt[b * TT + u];
  const int r = t_id / WW, c = t_id % WW;
  const float* __restrict__ row = logits + (size_t)bt * VV;
  const float m   = rowmax[bt];
  const float inv = 1.0f / rowsum[bt];

  float acc = 0.0f;
#pragma unroll 7
  for (int k = 0; k < NK; ++k) {
    const int rn = r + s_dy[k], cn = c + s_dx[k];
    if (rn >= 0 && rn < HH && cn >= 0 && cn < WW) {
      acc += __expf(row[rn * WW + cn] - m) * s_kw[k];
    }
  }
  U[bt * TT + u] = fmaxf(acc * inv, 1e-12f);
}

// ---------------------------------------------------------------------------
// Kernel 3: one block per (b, n). Build K = exp(S/eps) in LDS (zero-padded to
// 96x96), run 30 Sinkhorn iterations with v_wmma_f32_16x16x4_f32 matvecs
// (vector broadcast across the 16 N columns), then q = sum(a.K.b.S)/I and
// atomicAdd(-wn*log(q)/B).
// 192 threads = 6 waves; wave w owns output row-tile [16w, 16w+16).
// ---------------------------------------------------------------------------
__global__ __launch_bounds__(192) void sinkhorn_kernel(
    const float* __restrict__ U, const float* __restrict__ ngw,
    float* __restrict__ out) {
  __shared__ float Klds[96 * PITCH];
  __shared__ float veca[96];
  __shared__ float vecb[96];
  __shared__ float yz[96];
  __shared__ float red[192];

  const int blk = blockIdx.x;
  const int b   = blk / NGRAMS;
  const int n   = (blk % NGRAMS) + 1;
  const int I   = TT - n + 1;
  const int tid = threadIdx.x;
  const float* __restrict__ Ub = U + (size_t)b * TT * TT;

  // Build K (and implicitly S = 0.1*log(K)); zero pad outside IxI.
  for (int e = tid; e < 96 * 96; e += 192) {
    const int i = e / 96, j = e % 96;
    float kv = 0.0f;
    if (i < I && j < I) {
      float s = Ub[i * TT + j];
      for (int k = 1; k < n; ++k) s *= Ub[(i + k) * TT + (j + k)];
      s = fmaxf(s, 1e-12f);
      kv = __expf(s * 10.0f);   // 1/EPS = 10
    }
    Klds[i * PITCH + j] = kv;
  }
  if (tid < 96) vecb[tid] = (tid < I) ? 1.0f : 0.0f;
  __syncthreads();

  const float muv  = 1.0f / (float)I;   // mu = nu = 1/I
  const int   wave = tid >> 5;
  const int   lane = tid & 31;
  const int   lm   = lane & 15;         // M (or N) index within tile
  const int   kh   = lane >> 4;         // K-half select (A/B frag layout)
  const int   row0 = wave * 16;

  for (int it = 0; it < 30; ++it) {
    // ---- y = K @ b  (A = K row-tile, B = b broadcast over 16 columns) ----
    {
      v8f acc = {0.f,0.f,0.f,0.f,0.f,0.f,0.f,0.f};
      for (int jc = 0; jc < 24; ++jc) {
        const int col = jc * 4 + kh * 2;
        v2f af; af.x = Klds[(row0 + lm) * PITCH + col];
                af.y = Klds[(row0 + lm) * PITCH + col + 1];
        v2f bf; bf.x = vecb[col];
                bf.y = vecb[col + 1];
        acc = __builtin_amdgcn_wmma_f32_16x16x4_f32(
            false, af, false, bf, (short)0, acc, false, false);
      }
      if (lm == 0) {
#pragma unroll
        for (int rr = 0; rr < 8; ++rr) yz[row0 + rr + 8 * kh] = acc[rr];
      }
    }
    __syncthreads();
    if (tid < 96) veca[tid] = ((tid < I) ? muv : 0.0f) / fmaxf(yz[tid], 1e-30f);
    __syncthreads();

    // ---- z = K^T @ a  (A-frag reads K columns; lane-stride-1 LDS reads) ----
    {
      v8f acc = {0.f,0.f,0.f,0.f,0.f,0.f,0.f,0.f};
      for (int jc = 0; jc < 24; ++jc) {
        const int kb = jc * 4 + kh * 2;
        v2f af; af.x = Klds[kb * PITCH + row0 + lm];
                af.y = Klds[(kb + 1) * PITCH + row0 + lm];
        v2f bf; bf.x = veca[kb];
                bf.y = veca[kb + 1];
        acc = __builtin_amdgcn_wmma_f32_16x16x4_f32(
            false, af, false, bf, (short)0, acc, false, false);
      }
      if (lm == 0) {
#pragma unroll
        for (int rr = 0; rr < 8; ++rr) yz[row0 + rr + 8 * kh] = acc[rr];
      }
    }
    __syncthreads();
    if (tid < 96) vecb[tid] = ((tid < I) ? muv : 0.0f) / fmaxf(yz[tid], 1e-30f);
    __syncthreads();
  }

  // q = sum_ij a_i * K_ij * b_j * S_ij ; S = 0.1*log(K) (clip never binds)
  float part = 0.0f;
  for (int e = tid; e < 96 * 96; e += 192) {
    const int i = e / 96, j = e % 96;
    if (i < I && j < I) {
      const float kv = Klds[i * PITCH + j];
      const float sv = 0.1f * __logf(kv);
      part += veca[i] * vecb[j] * kv * sv;
    }
  }
  red[tid] = part;
  __syncthreads();
  if (tid < 64) red[tid] += red[tid + 64] + red[tid + 128];
  __syncthreads();
  for (int off = 32; off > 0; off >>= 1) {
    if (tid < off) red[tid] += red[tid + off];
    __syncthreads();
  }
  if (tid == 0) {
    const float q  = fmaxf(red[0] / (float)I, 1e-12f);
    const float wn = ngw[n - 1];
    atomicAdd(out, -wn * __logf(q) * (1.0f / (float)BB));
  }
}

// ---------------------------------------------------------------------------
extern "C" void kernel_launch(void* const* d_in, const int* in_sizes, int n_in,
                              void* d_out, int out_size, void* d_ws, size_t ws_size,
                              hipStream_t stream) {
  const float* logits = (const float*)d_in[0];   // (B,T,V) f32
  const float* kw     = (const float*)d_in[1];   // (49,)   f32
  const float* ngw    = (const float*)d_in[2];   // (5,)    f32
  const int*   tgt    = (const int*)  d_in[3];   // (B,T)   i32
  const int*   offs   = (const int*)  d_in[4];   // (49,2)  i32
  float* out = (float*)d_out;

  float* rowmax = (float*)d_ws;              // 1536
  float* rowsum = rowmax + ROWS;             // 1536
  float* U      = rowsum + ROWS;             // 16*96*96 = 147456

  zero_out_kernel<<<1, 32, 0, stream>>>(out);
  row_stats_kernel<<<ROWS, 256, 0, stream>>>(logits, rowmax, rowsum);
  compute_u_kernel<<<ROWS, 96, 0, stream>>>(logits, kw, tgt, offs, rowmax, rowsum, U);
  sinkhorn_kernel<<<BB * NGRAMS, 192, 0, stream>>>(U, ngw, out);
}